// Encoder_89489938580185
// MI455X (gfx1250) — compile-verified
//
#include <hip/hip_runtime.h>

#define FEAT   128    // feature dim
#define K2     256    // 2*FEAT (GEMM K)
#define EMB    128    // embed dim (GEMM M)
#define NSAMP  25     // sampled neighbors
#define TCOLS  128    // batch columns per workgroup
#define PITCH  260    // padded LDS row pitch in floats (bank-conflict-free)

typedef __attribute__((ext_vector_type(2))) float    v2f;
typedef __attribute__((ext_vector_type(4))) float    v4f;
typedef __attribute__((ext_vector_type(8))) float    v8f;
typedef __attribute__((ext_vector_type(4))) unsigned v4u;
typedef __attribute__((ext_vector_type(4))) int      v4i;
typedef __attribute__((ext_vector_type(8))) int      v8i;

#if defined(__has_builtin)
#if __has_builtin(__builtin_amdgcn_tensor_load_to_lds) && __has_builtin(__builtin_amdgcn_s_wait_tensorcnt)
#define USE_TDM 1
#endif
#endif
#ifndef USE_TDM
#define USE_TDM 0
#endif

__launch_bounds__(256, 1)
__global__ void sage_encoder_wmma(const int* __restrict__ nodes,
                                  const int* __restrict__ neigh,
                                  const float* __restrict__ feat,
                                  const float* __restrict__ weight,
                                  float* __restrict__ out,
                                  int Btot)
{
    __shared__ float lds_w[EMB   * PITCH];   // ~130 KB: W, padded rows (TDM pads in HW)
    __shared__ float lds_c[TCOLS * PITCH];   // ~130 KB: combined tile, padded rows

    const int tid  = threadIdx.x;
    const int wave = tid >> 5;
    const int lane = tid & 31;
    const int col0 = blockIdx.x * TCOLS;

#if USE_TDM
    // ---------------- Phase 0: TDM async-load weight [128 x 256] f32 -> padded LDS ----
    // D# descriptor per CDNA5 ISA.
    // Padding: pad_interval=7 (every 256 DWORDs) pad_amount=3 (4 DWORDs) => 260-float pitch.
    if (wave == 0) {
        const unsigned long long wap = (unsigned long long)weight;
        // Generic LDS pointer: low 32 bits == LDS byte offset (aperture high half).
        const unsigned lds_addr = (unsigned)(unsigned long long)(void*)lds_w;

        v4u g0;
        g0[0] = 1u;                                   // count=1, is_restore=0, gather off
        g0[1] = lds_addr;                             // lds_addr
        g0[2] = (unsigned)(wap & 0xFFFFFFFFull);      // global_addr[31:0]
        g0[3] = (unsigned)((wap >> 32) & 0x1FFFFFFu)  // global_addr[56:32]
              | 0x80000000u;                          // type=2 ("image")

        v8i g1;
        g1[0] = (int)0x07D20000u;   // wg_mask=0, data_size=2(4B), pad_en=1, pad_int=7, pad_amt=3
        g1[1] = (int)(256u << 16);  // tensor_dim0 = 256 (bits 79:48 -> low 16 here)
        g1[2] = (int)(128u << 16);  // tensor_dim1 = 128 (bits 111:80 -> low 16 here)
        g1[3] = (int)(256u << 16);  // tile_dim0   = 256 (bits 127:112)
        g1[4] = 128;                // tile_dim1   = 128, tile_dim2 = 0
        g1[5] = 256;                // tensor_dim0_stride[31:0] = 256
        g1[6] = (int)0x80000000u;   // tensor_dim1_stride[15:0]=32768 at bits[31:16]
        g1[7] = 0;                  // tensor_dim1_stride[47:16] = 0

        v4i g2; g2[0] = 1; g2[1] = 1; g2[2] = 0; g2[3] = 0;          // dim2=dim3=1 (unused)
        v4i g3; g3[0] = 0; g3[1] = (int)(1u << 16); g3[2] = 0; g3[3] = 0; // dim4=1 (unused)

#if defined(__clang_major__) && __clang_major__ >= 23
        v8i gz; gz[0]=0; gz[1]=0; gz[2]=0; gz[3]=0; gz[4]=0; gz[5]=0; gz[6]=0; gz[7]=0;
        __builtin_amdgcn_tensor_load_to_lds(g0, g1, g2, g3, gz, 0);
#else
        __builtin_amdgcn_tensor_load_to_lds(g0, g1, g2, g3, 0);
#endif
    }
#else
    // ---------------- Phase 0 (fallback): VMEM preload weight into padded LDS --------
    for (int idx = tid; idx < EMB * (K2 / 4); idx += 256) {
        const int e  = idx >> 6;
        const int k4 = idx & 63;
        v4f w = *(const v4f*)(weight + (size_t)e * K2 + k4 * 4);
        *(v4f*)(lds_w + e * PITCH + k4 * 4) = w;
    }
#endif

    // ---------------- Phase 1: gather + mean -> combined tile in LDS ----------------
    // wave w owns local columns [16w, 16w+16). Each lane holds 4 of the 128 feats.
    const float* featL = feat + lane * 4;
    for (int c = 0; c < 16; ++c) {
        const int colLocal = wave * 16 + c;
        const int col      = col0 + colLocal;
        v4f selfF = {0.f, 0.f, 0.f, 0.f};
        v4f meanF = {0.f, 0.f, 0.f, 0.f};
        if (col < Btot) {
            const long long node = nodes[col];
            selfF = *(const v4f*)(featL + node * (long long)FEAT);
            const int* nb = neigh + (long long)col * NSAMP;
            v4f a = {0.f, 0.f, 0.f, 0.f};
            #pragma unroll 5
            for (int j = 0; j < NSAMP; ++j) {
                const long long nn = nb[j];
                v4f f = *(const v4f*)(featL + nn * (long long)FEAT);
                a.x += f.x; a.y += f.y; a.z += f.z; a.w += f.w;
            }
            const float inv = 1.0f / (float)NSAMP;
            meanF.x = a.x * inv; meanF.y = a.y * inv;
            meanF.z = a.z * inv; meanF.w = a.w * inv;
        }
        // combined[colLocal][0:128] = self, [128:256] = mean(neigh)
        *(v4f*)(lds_c + colLocal * PITCH +        lane * 4) = selfF;
        *(v4f*)(lds_c + colLocal * PITCH + FEAT + lane * 4) = meanF;
    }

#if USE_TDM
    if (wave == 0) {
        __builtin_amdgcn_s_wait_tensorcnt(0);   // wave 0 owns TENSORcnt
    }
#endif
    __syncthreads();

    // ---------------- Phase 2: D = relu(W x combined^T), 16x16x4 f32 WMMA ----------------
    // A 16x4 f32 layout: lanes 0-15 hold row M=l, {K=k, k+1}; lanes 16-31 hold {K=k+2, k+3}.
    // B  4x16 f32 layout (mirrored): lane half selects K pair, N = lane&15.
    const int half = lane >> 4;      // 0: K lo pair, 1: K hi pair
    const int l15  = lane & 15;

    const float* aBase = lds_w + l15 * PITCH + 2 * half;                 // W row l15 (+16m)
    const float* bBase = lds_c + (wave * 16 + l15) * PITCH + 2 * half;   // combined col n

    v8f acc[8];
    #pragma unroll
    for (int m = 0; m < 8; ++m) acc[m] = (v8f){0.f,0.f,0.f,0.f,0.f,0.f,0.f,0.f};

    for (int k = 0; k < K2; k += 4) {
        v2f b = *(const v2f*)(bBase + k);
        #pragma unroll
        for (int m = 0; m < 8; ++m) {
            v2f a = *(const v2f*)(aBase + m * 16 * PITCH + k);
            acc[m] = __builtin_amdgcn_wmma_f32_16x16x4_f32(
                false, a, false, b, (short)0, acc[m], false, false);
        }
    }

    // ---------------- Epilogue: relu + store ----------------
    // C/D layout: VGPR r -> (M = r, N = lane) for lanes 0-15; (M = r+8, N = lane-16) hi half.
    const int col     = col0 + wave * 16 + l15;
    const int rowBase = half * 8;
    if (col < Btot) {
        #pragma unroll
        for (int m = 0; m < 8; ++m) {
            #pragma unroll
            for (int r = 0; r < 8; ++r) {
                float v = acc[m][r];
                v = v > 0.f ? v : 0.f;
                out[(size_t)(m * 16 + rowBase + r) * (size_t)Btot + col] = v;
            }
        }
    }
}

extern "C" void kernel_launch(void* const* d_in, const int* in_sizes, int n_in,
                              void* d_out, int out_size, void* d_ws, size_t ws_size,
                              hipStream_t stream) {
    const int*   nodes  = (const int*)d_in[0];
    const int*   neigh  = (const int*)d_in[1];
    const float* feat   = (const float*)d_in[2];
    const float* weight = (const float*)d_in[3];
    float*       out    = (float*)d_out;

    const int Btot = in_sizes[0];
    const int grid = (Btot + TCOLS - 1) / TCOLS;
    sage_encoder_wmma<<<grid, 256, 0, stream>>>(nodes, neigh, feat, weight, out, Btot);
}